// Yolo_loss_22170621182694
// MI455X (gfx1250) — compile-verified
//
#include <hip/hip_runtime.h>
#include <hip/hip_bf16.h>
#include <cstdint>
#include <cstddef>

#define B_BATCH 16
#define NMAX    60
#define NCLS    80

typedef float v2f __attribute__((ext_vector_type(2)));
typedef float v8f __attribute__((ext_vector_type(8)));

__device__ __constant__ float d_ANCH[18] = {
  12.f,16.f, 19.f,36.f, 40.f,28.f, 36.f,75.f, 76.f,55.f,
  72.f,146.f, 142.f,110.f, 192.f,243.f, 459.f,401.f };

__device__ __forceinline__ float slogf_(float x){ return fmaxf(logf(x), -100.0f); }
__device__ __forceinline__ float bcef_(float p, float t){
  return -(t*slogf_(p) + (1.0f - t)*slogf_(1.0f - p));
}
__device__ __forceinline__ float sigm_(float x){ return 1.0f/(1.0f + expf(-x)); }

// Deterministic wave32 sum using V_WMMA_F32_16X16X4_F32:
// A = 16x4 ones, B = 4x16 with our 32 lane values in the first B VGPR (other
// VGPR zero). D[m][n] = colsum(B)[n] = v[n] + v[n+16]; finish 16->1 with
// xor-shuffles. Result valid (and equal) in all lanes.
__device__ __forceinline__ float wave_sum32(float v){
  v2f a; a.x = 1.0f; a.y = 1.0f;
  v2f bm; bm.x = v; bm.y = 0.0f;
  v8f c = {};
  v8f d = __builtin_amdgcn_wmma_f32_16x16x4_f32(false, a, false, bm,
                                                (short)0, c, false, false);
  float s = d[0];
  s += __shfl_xor(s, 1, 32);
  s += __shfl_xor(s, 2, 32);
  s += __shfl_xor(s, 4, 32);
  s += __shfl_xor(s, 8, 32);
  return s;
}

// ---------------------------------------------------------------------------
// Pass 1: per (level, b, n) label records. 12 floats each:
//  [0..3] target box tl/br (grid units), [4] area,
//  [5] frac_x, [6] frac_y, [7] lw, [8] lh, [9] scale, [10] cls,
//  [11] match code = (best_n*FS + tj)*FS + ti  (or -1 if no match), as int bits
// ---------------------------------------------------------------------------
__global__ void prep_kernel(const float* __restrict__ labels,
                            float* __restrict__ rec){
  const int b = blockIdx.x;
  const int n = threadIdx.x;
  if (n >= NMAX) return;
  const float* lb = labels + (size_t)(b*NMAX + n)*5;
  const float x1 = lb[0], y1 = lb[1], x2 = lb[2], y2 = lb[3], clsf = lb[4];
  const bool valid = (x1 + y1 + x2 + y2 + clsf) > 0.0f;
  #pragma unroll
  for (int L = 0; L < 3; ++L){
    const float s  = (L==0) ? 8.f : ((L==1) ? 16.f : 32.f);
    const int   fs = (L==0) ? 76  : ((L==1) ? 38   : 19);
    const float inv = 1.0f/s;
    const float tx = (x2+x1)*0.5f*inv;
    const float ty = (y2+y1)*0.5f*inv;
    const float tw = (x2-x1)*inv;
    const float th = (y2-y1)*inv;
    // argmax over 9 anchor IoUs (first max wins, like jnp.argmax)
    float bestiou = -1.0f; int best = 0;
    #pragma unroll
    for (int a2 = 0; a2 < 9; ++a2){
      const float aw = d_ANCH[2*a2]*inv, ah = d_ANCH[2*a2+1]*inv;
      const float it = fminf(tw,aw)*fminf(th,ah);
      const float un = tw*th + aw*ah - it;
      const float io = it/un;
      if (io > bestiou){ bestiou = io; best = a2; }
    }
    const int  bn = best % 3;
    const bool wm = valid && ((best/3) == L);
    const int  ti = (int)tx, tj = (int)ty;
    const float maw = d_ANCH[2*(3*L+bn)]*inv;
    const float mah = d_ANCH[2*(3*L+bn)+1]*inv;
    const float fw = (float)fs;
    float* r = rec + (size_t)((L*B_BATCH + b)*NMAX + n)*12;
    r[0] = tx - 0.5f*tw;  r[1] = ty - 0.5f*th;
    r[2] = tx + 0.5f*tw;  r[3] = ty + 0.5f*th;
    r[4] = tw*th;
    r[5] = tx - floorf(tx);
    r[6] = ty - floorf(ty);
    r[7] = logf(tw/maw + 1e-16f);
    r[8] = logf(th/mah + 1e-16f);
    r[9] = sqrtf(2.0f - tw*th/(fw*fw));
    r[10] = clsf;
    const int code = wm ? ((bn*fs + tj)*fs + ti) : -1;
    r[11] = __int_as_float(code);
  }
}

// ---------------------------------------------------------------------------
// Pass 2: per-cell loss. One thread per grid cell; reads only channels 0..4
// for every cell (coalesced per channel plane), and channels 5..84 only at
// the rare matched cells. IoU loop runs out of LDS: one ds_load_b128 (box)
// + one ds_load_b32 (area) + one ds_load_b32 (code) per iteration, all
// uniform-address broadcasts.
// ---------------------------------------------------------------------------
template<int FS, int LEVEL>
__global__ void __launch_bounds__(256)
loss_kernel(const float* __restrict__ xin, const float* __restrict__ rec,
            float* __restrict__ partial){
  constexpr int F2  = FS*FS;
  constexpr int BPI = (F2 + 255)/256;      // blocks per (b, anchor) image
  const int blk   = blockIdx.x;
  const int img   = blk / BPI;             // b*3 + a
  const int chunk = blk - img*BPI;
  const int b = img / 3;
  const int a = img - b*3;
  const int t = threadIdx.x;
  const int cell = chunk*256 + t;

  __shared__ float4 s_box[NMAX];           // target box tl/br as one b128
  __shared__ float  s_area[NMAX];          // target area
  __shared__ float  s_mt[NMAX*6];          // fracx,fracy,lw,lh,scale,cls
  __shared__ int    s_code[NMAX];          // match code (-1 if none)
  {
    const float* r = rec + (size_t)((LEVEL*B_BATCH + b)*NMAX)*12;
    float* boxf = (float*)s_box;
    for (int k = t; k < NMAX*12; k += 256){
      const int n = k/12, f = k - n*12;
      const float v = r[k];
      if (f < 4)        boxf[n*4+f]       = v;
      else if (f == 4)  s_area[n]         = v;
      else if (f < 11)  s_mt[n*6 + (f-5)] = v;
      else              s_code[n]         = __float_as_int(v);
    }
  }
  __syncthreads();

  float axy = 0.f, awh = 0.f, aob = 0.f, acl = 0.f, al2 = 0.f;

  if (cell < F2){
    const int i = cell % FS;
    const int j = cell / FS;
    constexpr float stride = (LEVEL==0) ? 8.f : ((LEVEL==1) ? 16.f : 32.f);
    const float* base = xin + (size_t)(b*3 + a)*85*F2 + cell;
    const float o0 = base[0];
    const float o1 = base[(size_t)F2];
    const float o2 = base[(size_t)2*F2];
    const float o3 = base[(size_t)3*F2];
    const float o4 = base[(size_t)4*F2];
    const float sx = sigm_(o0);
    const float sy = sigm_(o1);
    const float anw = d_ANCH[2*(3*LEVEL + a)]   / stride;
    const float anh = d_ANCH[2*(3*LEVEL + a)+1] / stride;
    const float pw = expf(o2)*anw;
    const float ph = expf(o3)*anh;
    const float px = sx + (float)i;
    const float py = sy + (float)j;
    const float hw = 0.5f*pw, hh = 0.5f*ph;
    const float ptlx = px - hw, ptly = py - hh;
    const float pbrx = px + hw, pbry = py + hh;
    const float areap = pw*ph;

    float best = 0.f;
    int m = -1, hm = 0;
    const int myCode = (a*FS + j)*FS + i;
    #pragma unroll 4
    for (int n = 0; n < NMAX; ++n){
      const float4 tb = s_box[n];
      const float sumA = areap + s_area[n];
      const float iw = fmaxf(fminf(pbrx,tb.z) - fmaxf(ptlx,tb.x), 0.f);
      const float ih = fmaxf(fminf(pbry,tb.w) - fmaxf(ptly,tb.y), 0.f);
      const float inter = iw*ih;
      const float iou = inter / (sumA - inter);
      best = fmaxf(best, iou);
      const int cn = s_code[n];
      hm |= (cn >= 0) ? 1 : 0;
      if (cn == myCode) m = n;   // ascending n -> last writer wins
    }

    float obj_mask = ((best > 0.5f) && hm) ? 0.f : 1.f;
    if (m >= 0) obj_mask = 1.f;
    const float p4 = sigm_(o4);
    if (obj_mask > 0.f){
      const float tobj = (m >= 0) ? 1.f : 0.f;
      aob += -slogf_((m >= 0) ? p4 : (1.f - p4));   // BCE(p4, tobj)
      const float d = p4 - tobj;
      al2 += d*d;
    }
    if (m >= 0){
      const float* mt = &s_mt[m*6];
      const float scale = mt[4];
      const float w2 = scale*scale;
      axy += w2*(bcef_(sx, mt[0]) + bcef_(sy, mt[1]));
      const float d0 = sx - mt[0], d1 = sy - mt[1];
      al2 += d0*d0 + d1*d1;
      const float dw0 = (o2 - mt[2])*scale;
      const float dw1 = (o3 - mt[3])*scale;
      awh += 0.5f*(dw0*dw0 + dw1*dw1);
      al2 += dw0*dw0 + dw1*dw1;
      const int cls = (int)mt[5];
      for (int c = 0; c < NCLS; ++c){
        const float p  = sigm_(base[(size_t)(5+c)*F2]);
        const float tc = (c == cls) ? 1.f : 0.f;
        acl += -slogf_((c == cls) ? p : (1.f - p));
        const float dd = p - tc;
        al2 += dd*dd;
      }
    }
  }

  // Deterministic block reduction: WMMA wave sums -> LDS -> thread 0.
  __shared__ float s_w[8][5];
  const int wave = t >> 5, lane = t & 31;
  const float r0 = wave_sum32(axy);
  const float r1 = wave_sum32(awh);
  const float r2 = wave_sum32(aob);
  const float r3 = wave_sum32(acl);
  const float r4 = wave_sum32(al2);
  if (lane == 0){
    s_w[wave][0]=r0; s_w[wave][1]=r1; s_w[wave][2]=r2; s_w[wave][3]=r3; s_w[wave][4]=r4;
  }
  __syncthreads();
  if (t == 0){
    float s0=0,s1=0,s2=0,s3=0,s4=0;
    #pragma unroll
    for (int w = 0; w < 8; ++w){
      s0+=s_w[w][0]; s1+=s_w[w][1]; s2+=s_w[w][2]; s3+=s_w[w][3]; s4+=s_w[w][4];
    }
    float* p = partial + (size_t)blk*5;
    p[0]=s0; p[1]=s1; p[2]=s2; p[3]=s3; p[4]=s4;
  }
}

// ---------------------------------------------------------------------------
// Pass 3: fold per-block partials into the 6 outputs, deterministically.
// ---------------------------------------------------------------------------
__global__ void __launch_bounds__(256)
final_kernel(const float* __restrict__ partial, float* __restrict__ out,
             int nblocks){
  float a0=0,a1=0,a2=0,a3=0,a4=0;
  for (int ib = threadIdx.x; ib < nblocks; ib += 256){
    const float* p = partial + (size_t)ib*5;
    a0+=p[0]; a1+=p[1]; a2+=p[2]; a3+=p[3]; a4+=p[4];
  }
  __shared__ float s_w[8][5];
  const int wave = threadIdx.x >> 5, lane = threadIdx.x & 31;
  const float r0 = wave_sum32(a0);
  const float r1 = wave_sum32(a1);
  const float r2 = wave_sum32(a2);
  const float r3 = wave_sum32(a3);
  const float r4 = wave_sum32(a4);
  if (lane == 0){
    s_w[wave][0]=r0; s_w[wave][1]=r1; s_w[wave][2]=r2; s_w[wave][3]=r3; s_w[wave][4]=r4;
  }
  __syncthreads();
  if (threadIdx.x == 0){
    float lxy=0,lwh=0,lobj=0,lcls=0,ll2=0;
    #pragma unroll
    for (int w = 0; w < 8; ++w){
      lxy+=s_w[w][0]; lwh+=s_w[w][1]; lobj+=s_w[w][2]; lcls+=s_w[w][3]; ll2+=s_w[w][4];
    }
    out[0] = lxy + lwh + lobj + lcls;
    out[1] = lxy; out[2] = lwh; out[3] = lobj; out[4] = lcls; out[5] = ll2;
  }
}

extern "C" void kernel_launch(void* const* d_in, const int* in_sizes, int n_in,
                              void* d_out, int out_size, void* d_ws, size_t ws_size,
                              hipStream_t stream){
  const float* xin0   = (const float*)d_in[0];
  const float* xin1   = (const float*)d_in[1];
  const float* xin2   = (const float*)d_in[2];
  const float* labels = (const float*)d_in[3];

  float* rec     = (float*)d_ws;                    // 3*16*60*12 = 34560 floats
  float* partial = rec + (size_t)3*B_BATCH*NMAX*12; // 1488*5 floats

  constexpr int NB0 = B_BATCH*3*((76*76 + 255)/256); // 1104
  constexpr int NB1 = B_BATCH*3*((38*38 + 255)/256); // 288
  constexpr int NB2 = B_BATCH*3*((19*19 + 255)/256); // 96

  prep_kernel<<<B_BATCH, 64, 0, stream>>>(labels, rec);
  loss_kernel<76,0><<<NB0, 256, 0, stream>>>(xin0, rec, partial);
  loss_kernel<38,1><<<NB1, 256, 0, stream>>>(xin1, rec, partial + (size_t)NB0*5);
  loss_kernel<19,2><<<NB2, 256, 0, stream>>>(xin2, rec, partial + (size_t)(NB0+NB1)*5);
  final_kernel<<<1, 256, 0, stream>>>(partial, (float*)d_out, NB0+NB1+NB2);
}